// arrayPC_55963423867092
// MI455X (gfx1250) — compile-verified
//
#include <hip/hip_runtime.h>
#include <stdint.h>

// ---------------------------------------------------------------------------
// Problem constants (from reference): B=8192 rows, N=1024 steps, K=32 channels
// ---------------------------------------------------------------------------
#define BB       8192
#define NN       1024
#define KK       32
#define NWROWS   1023          // W rows actually used (N-1)
#define ROWF     62            // floats per Wlog row: (K-1)*2
#define CHUNK    64            // Wlog rows staged per LDS buffer
#define NCHUNK   16            // ceil(1023/64) (row 1023 is padding)
#define CH_F     (CHUNK*ROWF)  // floats per chunk = 3968
#define CH_F4    (CH_F/4)      // float4s per chunk = 992

#define LOG_EPS  (-34.538776394910684f)   // log(float32(1e-15))

// ---------------------------------------------------------------------------
// CDNA5 async global->LDS path (guarded; falls back to ld + ds_store)
// ---------------------------------------------------------------------------
#if defined(__HIP_DEVICE_COMPILE__) && defined(__has_builtin)
#  if __has_builtin(__builtin_amdgcn_global_load_async_to_lds_b128)
#    define USE_ASYNC 1
#  else
#    define USE_ASYNC 0
#  endif
#else
#  define USE_ASYNC 0
#endif

#if USE_ASYNC
typedef __attribute__((__vector_size__(16))) int v4i;
typedef __attribute__((address_space(1))) v4i gv4i;   // global int4
typedef __attribute__((address_space(3))) v4i lv4i;   // LDS int4
#endif

__device__ __forceinline__ void wait_async_zero() {
#if USE_ASYNC
#  if __has_builtin(__builtin_amdgcn_s_wait_asynccnt)
  __builtin_amdgcn_s_wait_asynccnt(0);
#  else
  asm volatile("s_wait_asynccnt 0x0" ::: "memory");
#  endif
#endif
}

// Stage CH_F floats (CHUNK Wlog rows, contiguous) from global into LDS.
__device__ __forceinline__ void stage_chunk(const float* __restrict__ gsrc,
                                            float* lds_dst, int tid) {
#if USE_ASYNC
  gv4i* g = (gv4i*)const_cast<float*>(gsrc);
  lv4i* l = (lv4i*)lds_dst;
#pragma unroll
  for (int k = 0; k < 4; ++k) {
    int idx = tid + k * 256;               // float4 index
    if (idx < CH_F4)
      __builtin_amdgcn_global_load_async_to_lds_b128(g + idx, l + idx, 0, 0);
  }
#else
#pragma unroll
  for (int k = 0; k < 4; ++k) {
    int idx = tid + k * 256;
    if (idx < CH_F4) {
      float4 v = reinterpret_cast<const float4*>(gsrc)[idx];
      reinterpret_cast<float4*>(lds_dst)[idx] = v;
    }
  }
#endif
}

// ---------------------------------------------------------------------------
// Prep 1: Wlog[i][j][c] = log(softmax(W[i,j,:])[c]*adj1 + adj2 + eps)
//         (band mask applied to rows i<30), plus end = log_softmax(endW).
// ---------------------------------------------------------------------------
__global__ __launch_bounds__(256) void prep_w_kernel(
    const float* __restrict__ W, const float* __restrict__ endW,
    float* __restrict__ wlog, float* __restrict__ endls) {
  int idx = blockIdx.x * 256 + threadIdx.x;   // over 1023*31 (i,j) pairs
  if (idx < NWROWS * (KK - 1)) {
    int i = idx / (KK - 1);
    int j = idx - i * (KK - 1);
    float2 w = reinterpret_cast<const float2*>(W)[i * (KK - 1) + j];
    float m  = fmaxf(w.x, w.y);
    float p0 = expf(w.x - m), p1 = expf(w.y - m);
    float inv = 1.0f / (p0 + p1);
    float sm0 = p0 * inv, sm1 = p1 * inv;
    float o0, o1;
    if (i < KK - 2) {                 // masked rows 0..29
      if (j <= i)          { o0 = logf(sm0 + 1e-15f); o1 = logf(sm1 + 1e-15f); }
      else if (j == i + 1) { o0 = 0.0f;               o1 = LOG_EPS;            }
      else                 { o0 = LOG_EPS;            o1 = LOG_EPS;            }
    } else {
      o0 = logf(sm0 + 1e-15f); o1 = logf(sm1 + 1e-15f);
    }
    wlog[i * ROWF + 2 * j]     = o0;
    wlog[i * ROWF + 2 * j + 1] = o1;
  }
  if (idx == 0) {                     // log_softmax(endW) over K=32
    float m = -3.4e38f;
    for (int k = 0; k < KK; ++k) m = fmaxf(m, endW[k]);
    float s = 0.0f;
    for (int k = 0; k < KK; ++k) s += expf(endW[k] - m);
    float lse = m + logf(s);
    for (int k = 0; k < KK; ++k) endls[k] = endW[k] - lse;
  }
}

// ---------------------------------------------------------------------------
// Prep 2: bit-pack x (bool bytes) -> one uint32 per 32 steps, LSB-first.
// ---------------------------------------------------------------------------
__global__ __launch_bounds__(256) void pack_x_kernel(
    const unsigned char* __restrict__ x, uint32_t* __restrict__ xp) {
  int idx = blockIdx.x * 256 + threadIdx.x;   // over B * (N/32)
  int r = idx >> 5;
  int w = idx & 31;
  const uint32_t* p = reinterpret_cast<const uint32_t*>(x + r * NN + w * 32);
  uint32_t bits = 0;
#pragma unroll
  for (int m = 0; m < 8; ++m) {
    uint32_t u = p[m];
#pragma unroll
    for (int b = 0; b < 4; ++b)
      bits |= ((u >> (8 * b)) & 1u) << (m * 4 + b);
  }
  xp[idx] = bits;
}

// ---------------------------------------------------------------------------
// Main scan: lane j of each wave owns channel j of one batch row.
// 8 waves/block (8 rows), 1024 blocks. Wlog double-buffered in LDS.
// ---------------------------------------------------------------------------
__global__ __launch_bounds__(256) void scan_kernel(
    const uint32_t* __restrict__ xp, const float* __restrict__ wlog,
    const float* __restrict__ endls, float* __restrict__ out) {
  __shared__ float wsm[2 * CH_F];             // 31,744 bytes

  const int tid  = threadIdx.x;
  const int lane = tid & 31;
  const int wid  = tid >> 5;
  const int row  = blockIdx.x * 8 + wid;

  const int jm    = (lane > 0) ? lane - 1 : 0;  // source channel for 'a' path
  const int bpidx = jm << 2;                    // ds_bpermute byte index
  const uint32_t* xprow = xp + (row << 5);

  // ---- step 0 (initial f) ----
  uint32_t xw = xprow[0];
  int bit0 = (int)(xw & 1u);
  float f;
  if (lane == 0)      f = bit0 ? LOG_EPS : 0.0f;  // log(cumprod + eps)
  else if (lane == 1) f = bit0 ? 0.0f : LOG_EPS;  // log(x0 + eps)
  else                f = LOG_EPS;
  int seen = bit0;

  stage_chunk(wlog, wsm, tid);                  // chunk 0 -> buffer 0

  int i = 1;                                    // scan step index (x column)
  for (int c = 0; c < NCHUNK; ++c) {
    wait_async_zero();
    __syncthreads();                            // chunk c resident for all
    if (c + 1 < NCHUNK)
      stage_chunk(wlog + (c + 1) * CH_F, wsm + ((c + 1) & 1) * CH_F, tid);

    const float* wrow = wsm + (c & 1) * CH_F;
    const int nsteps = (c == NCHUNK - 1) ? (NWROWS - c * CHUNK) : CHUNK;

#pragma unroll 4
    for (int s = 0; s < nsteps; ++s, ++i) {
      if ((i & 31) == 0) xw = xprow[i >> 5];
      int bit = (int)((xw >> (i & 31)) & 1u);
      float lx  = bit ? 0.0f : LOG_EPS;
      float lxb = bit ? LOG_EPS : 0.0f;
      seen |= bit;
      float lf0 = seen ? LOG_EPS : 0.0f;

      float2 w = *reinterpret_cast<const float2*>(wrow + s * ROWF + (jm << 1));
      float fup = __int_as_float(
          __builtin_amdgcn_ds_bpermute(bpidx, __float_as_int(f)));

      float a  = w.x + fup + lx;                // wl[j-1,0] + f[j-1] + logx
      float b  = w.y + f   + lxb;               // wl[j-1,1] + f[j]   + logxbar
      float mx = fmaxf(a, b);
      float mn = fminf(a, b);
      float r  = mx + __logf(1.0f + __expf(mn - mx));   // logaddexp
      f = (lane == 0) ? lf0 : r;
    }
    // next iteration's top barrier guards buffer reuse
  }

  // ---- out[row] = max_j(end[j] + f_last[j]) ----
  float v = endls[lane] + f;
#pragma unroll
  for (int off = 16; off > 0; off >>= 1)
    v = fmaxf(v, __shfl_xor(v, off, 32));
  if (lane == 0) out[row] = v;
}

// ---------------------------------------------------------------------------
// Host launcher
// ---------------------------------------------------------------------------
extern "C" void kernel_launch(void* const* d_in, const int* in_sizes, int n_in,
                              void* d_out, int out_size, void* d_ws,
                              size_t ws_size, hipStream_t stream) {
  (void)in_sizes; (void)n_in; (void)out_size; (void)ws_size;
  const unsigned char* x = (const unsigned char*)d_in[0];  // bool (B,N)
  const float* W    = (const float*)d_in[1];               // (1023,31,2)
  const float* endW = (const float*)d_in[2];               // (1,32)
  float* out = (float*)d_out;                              // (B,)

  // Workspace layout: wlog (1024 rows x 62 f, row 1023 = pad), endls(32), xp
  float* wlog  = (float*)d_ws;
  float* endls = wlog + 1024 * ROWF;
  uint32_t* xp = (uint32_t*)(endls + 32);

  prep_w_kernel<<<dim3((NWROWS * (KK - 1) + 255) / 256), dim3(256), 0, stream>>>(
      W, endW, wlog, endls);
  pack_x_kernel<<<dim3(BB * (NN / 32) / 256), dim3(256), 0, stream>>>(x, xp);
  scan_kernel<<<dim3(BB / 8), dim3(256), 0, stream>>>(xp, wlog, endls, out);
}